// SynergyGraph_42348377539154
// MI455X (gfx1250) — compile-verified
//
#include <hip/hip_runtime.h>

typedef __attribute__((ext_vector_type(2))) float v2f;
typedef __attribute__((ext_vector_type(4))) float v4f;
typedef __attribute__((ext_vector_type(8))) float v8f;

#define BATCH 16
#define SEQ   2048
#define DIM   64
#define TILE  128
#define LDS_STRIDE 68   // 64 + 4 floats pad: (4*row + col) mod 64 unique per half-wave -> no bank conflicts

__global__ __launch_bounds__(256)
void SynergyGraph_42348377539154_kernel(const int* __restrict__ seq,
                                        const float* __restrict__ weight,
                                        float* __restrict__ out)
{
    __shared__ float ldsA[TILE * LDS_STRIDE];
    __shared__ float ldsB[TILE * LDS_STRIDE];

    const int b  = blockIdx.z;
    const int i0 = blockIdx.y * TILE;
    const int j0 = blockIdx.x * TILE;

    const int t = threadIdx.x;              // 0..255
    const int* seq_b = seq + b * SEQ;

    // ---- Gather 128 i-rows and 128 j-rows of the embedding into LDS ----
    // 128 rows * 16 float4 = 2048 float4 chunks per block side; 256 threads -> 8 each.
    #pragma unroll
    for (int it = 0; it < 8; ++it) {
        int chunk = t + 256 * it;           // 0..2047
        int row   = chunk >> 4;             // 0..127
        int c4    = (chunk & 15) << 2;      // 0,4,...,60
        int ia = seq_b[i0 + row];
        v4f va = *(const v4f*)(weight + (size_t)ia * DIM + c4);
        *(v4f*)(&ldsA[row * LDS_STRIDE + c4]) = va;
        int ib = seq_b[j0 + row];
        v4f vb = *(const v4f*)(weight + (size_t)ib * DIM + c4);
        *(v4f*)(&ldsB[row * LDS_STRIDE + c4]) = vb;
    }
    __syncthreads();

    // ---- Wave tiling: 8 waves in a 4x2 grid; each wave -> 32x64 output, 2x4 WMMA tiles ----
    const int wave = t >> 5;                // 0..7
    const int lane = t & 31;
    const int wr   = wave >> 1;             // 0..3 : 32-row strip
    const int wc   = wave & 1;              // 0..1 : 64-col strip

    const int m16 = lane & 15;              // row/col within 16x16 fragment
    const int kh  = lane >> 4;              // K-pair select (K = k0 + 2*kh, +1)

    v8f acc[2][4] = {};

    const int arow0 = wr * 32;
    const int brow0 = wc * 64;

    for (int k0 = 0; k0 < DIM; k0 += 4) {
        const int kc = k0 + 2 * kh;
        v2f a[2], bm[4];
        #pragma unroll
        for (int ti = 0; ti < 2; ++ti)
            a[ti] = *(const v2f*)(&ldsA[(arow0 + ti * 16 + m16) * LDS_STRIDE + kc]);
        #pragma unroll
        for (int tj = 0; tj < 4; ++tj)
            bm[tj] = *(const v2f*)(&ldsB[(brow0 + tj * 16 + m16) * LDS_STRIDE + kc]);

        #pragma unroll
        for (int ti = 0; ti < 2; ++ti)
            #pragma unroll
            for (int tj = 0; tj < 4; ++tj)
                acc[ti][tj] = __builtin_amdgcn_wmma_f32_16x16x4_f32(
                    /*neg_a=*/false, a[ti],
                    /*neg_b=*/false, bm[tj],
                    /*c_mod=*/(short)0, acc[ti][tj],
                    /*reuse_a=*/false, /*reuse_b=*/false);
    }

    // ---- Store: D layout -> VGPR r: n = lane%16, m = r + 8*(lane>>4). Scale = 1/sqrt(64). ----
    const float scale = 0.125f;
    const int   n     = m16;
    const int   mbase = 8 * kh;

    float* outb = out + (size_t)b * SEQ * SEQ;
    #pragma unroll
    for (int ti = 0; ti < 2; ++ti) {
        #pragma unroll
        for (int tj = 0; tj < 4; ++tj) {
            #pragma unroll
            for (int r = 0; r < 8; ++r) {
                int row = i0 + arow0 + ti * 16 + mbase + r;
                int col = j0 + brow0 + tj * 16 + n;
                outb[(size_t)row * SEQ + col] = acc[ti][tj][r] * scale;
            }
        }
    }
}

extern "C" void kernel_launch(void* const* d_in, const int* in_sizes, int n_in,
                              void* d_out, int out_size, void* d_ws, size_t ws_size,
                              hipStream_t stream) {
    (void)in_sizes; (void)n_in; (void)out_size; (void)d_ws; (void)ws_size;
    const int*   seq    = (const int*)d_in[0];     // [16, 2048] int32
    const float* weight = (const float*)d_in[1];   // [32000, 64] f32
    float*       out    = (float*)d_out;           // [16, 2048, 2048] f32

    dim3 grid(SEQ / TILE, SEQ / TILE, BATCH);      // 16 x 16 x 16
    dim3 block(256);                               // 8 wave32s
    SynergyGraph_42348377539154_kernel<<<grid, block, 0, stream>>>(seq, weight, out);
}